// hinge_loss_1_13975823581920
// MI455X (gfx1250) — compile-verified
//
#include <hip/hip_runtime.h>
#include <hip/hip_bf16.h>

typedef __attribute__((ext_vector_type(2))) float v2f;
typedef __attribute__((ext_vector_type(8))) float v8f;

#define S_TOTAL 8192
#define TILE    128   // macro-tile edge per block
#define NTILES  (S_TOTAL / TILE)   // 64

// ---------------------------------------------------------------------------
// Kernel 1: gather sampled scores/labels into dense workspace arrays.
// ---------------------------------------------------------------------------
__global__ void __launch_bounds__(256)
gather_kernel(const float* __restrict__ probs,
              const float* __restrict__ targets,
              const int* __restrict__ idx,
              float* __restrict__ P,
              float* __restrict__ T,
              int s_count) {
    int s = blockIdx.x * blockDim.x + threadIdx.x;
    if (s < s_count) {
        int k = idx[s];
        P[s] = probs[k];
        T[s] = targets[k];
    }
}

// ---------------------------------------------------------------------------
// Kernel 2: pair sweep. Each block owns a 128(i) x 128(j) macro-tile of the
// pair matrix. 8 wave32 per block; wave w owns rows [i0+16w, i0+16w+16) and
// loops over eight 16-wide j sub-tiles. For each 16x16 sub-tile, the margin
// matrix  m[i][j] = 1 - p_i + p_j  is produced by a single
// V_WMMA_F32_16X16X4_F32 with the rank-2 construction:
//     A[i][:] = [1, -p_i, 0, 0]       (lanes 16-31 hold K=2,3 -> zeros)
//     B[:][j] = [1+p_j; 1; 0; 0]^T    (lanes 16-31 hold K=2,3 -> zeros)
// Mask (t_i > t_j) and relu are applied on the 8 accumulator VGPRs in VALU.
// ---------------------------------------------------------------------------
__global__ void __launch_bounds__(256)
pair_hinge_kernel(const float* __restrict__ P,
                  const float* __restrict__ T,
                  float* __restrict__ blockSums) {
    __shared__ float sPj[TILE];
    __shared__ float sTj[TILE];
    __shared__ float sPi[TILE];
    __shared__ float sTi[TILE];
    __shared__ float red[256];

    const int tid    = threadIdx.x;
    const int j_base = blockIdx.x * TILE;
    const int i_base = blockIdx.y * TILE;

    // Stage this block's P/T slices into LDS (all inner-loop reads hit LDS).
    if (tid < TILE) {
        sPj[tid] = P[j_base + tid];
        sTj[tid] = T[j_base + tid];
    } else {
        int t2 = tid - TILE;
        sPi[t2] = P[i_base + t2];
        sTi[t2] = T[i_base + t2];
    }
    __syncthreads();

    const int lane = tid & 31;
    const int wave = tid >> 5;
    const int li   = lane & 15;   // column / M index within half-wave
    const int hi   = lane >> 4;   // half-wave select (holds K=2,3 operand rows)
    const int iw   = wave * 16;   // wave's row offset inside the macro-tile

    // A-matrix operand: lanes 0-15 -> K=0 (=1.0), K=1 (=-p_i); lanes 16-31 -> 0.
    v2f a;
    a.x = (hi == 0) ? 1.0f        : 0.0f;
    a.y = (hi == 0) ? -sPi[iw + li] : 0.0f;

    // t_i for the 8 accumulator rows this lane sees (rows r+8*hi).
    float tir[8];
#pragma unroll
    for (int r = 0; r < 8; ++r) tir[r] = sTi[iw + 8 * hi + r];

    float acc = 0.0f;
#pragma unroll
    for (int jt = 0; jt < 8; ++jt) {
        const int j0 = jt * 16;
        const float pj = sPj[j0 + li];
        const float tj = sTj[j0 + li];

        // B-matrix operand: lanes 0-15 -> K=0 (=1+p_j), K=1 (=1.0); lanes 16-31 -> 0.
        v2f b;
        b.x = (hi == 0) ? (1.0f + pj) : 0.0f;
        b.y = (hi == 0) ? 1.0f        : 0.0f;

        v8f c = {};
        // D[m][n] = 1 + p_n - p_m   (16x16 tile, 8 accumulator VGPRs/lane)
        v8f d = __builtin_amdgcn_wmma_f32_16x16x4_f32(
            /*neg_a=*/false, a, /*neg_b=*/false, b,
            /*c_mod=*/(short)0, c, /*reuse_a=*/false, /*reuse_b=*/false);

#pragma unroll
        for (int r = 0; r < 8; ++r) {
            float h = fmaxf(d[r], 0.0f);          // relu(1 - p_i + p_j)
            acc += (tir[r] > tj) ? h : 0.0f;      // ordering mask
        }
    }

    // Deterministic fixed-order block reduction.
    red[tid] = acc;
    __syncthreads();
#pragma unroll
    for (int s = 128; s > 0; s >>= 1) {
        if (tid < s) red[tid] += red[tid + s];
        __syncthreads();
    }
    if (tid == 0) blockSums[blockIdx.y * gridDim.x + blockIdx.x] = red[0];
}

// ---------------------------------------------------------------------------
// Kernel 3: deterministic final reduction of the 4096 block partials in
// double precision (guards fp32 accumulation error on ~3e7 magnitude sum).
// ---------------------------------------------------------------------------
__global__ void __launch_bounds__(256)
final_reduce_kernel(const float* __restrict__ blockSums, int n,
                    float* __restrict__ out) {
    __shared__ double red[256];
    double s = 0.0;
    for (int i = threadIdx.x; i < n; i += 256) s += (double)blockSums[i];
    red[threadIdx.x] = s;
    __syncthreads();
#pragma unroll
    for (int k = 128; k > 0; k >>= 1) {
        if (threadIdx.x < k) red[threadIdx.x] += red[threadIdx.x + k];
        __syncthreads();
    }
    if (threadIdx.x == 0) out[0] = (float)red[0];
}

// ---------------------------------------------------------------------------
extern "C" void kernel_launch(void* const* d_in, const int* in_sizes, int n_in,
                              void* d_out, int out_size, void* d_ws, size_t ws_size,
                              hipStream_t stream) {
    (void)n_in; (void)out_size; (void)ws_size;

    const float* probs   = (const float*)d_in[0];
    const float* targets = (const float*)d_in[1];
    const int*   idx     = (const int*)d_in[2];
    const int    s_count = in_sizes[2];   // == S_TOTAL (8192)

    float* P         = (float*)d_ws;          // 8192 floats
    float* T         = P + S_TOTAL;           // 8192 floats
    float* blockSums = T + S_TOTAL;           // NTILES*NTILES = 4096 floats

    gather_kernel<<<(s_count + 255) / 256, 256, 0, stream>>>(
        probs, targets, idx, P, T, s_count);

    dim3 grid(NTILES, NTILES);
    pair_hinge_kernel<<<grid, 256, 0, stream>>>(P, T, blockSums);

    final_reduce_kernel<<<1, 256, 0, stream>>>(
        blockSums, NTILES * NTILES, (float*)d_out);
}